// GraphGenerator_15496242004546
// MI455X (gfx1250) — compile-verified
//
#include <hip/hip_runtime.h>

typedef __attribute__((ext_vector_type(2))) float v2f;
typedef __attribute__((ext_vector_type(8))) float v8f;

#define BATCH     64
#define LATENT    128
#define HID       256
#define MAX_NODES 512
#define NODE_FEAT 64
#define NF_N      (MAX_NODES * NODE_FEAT)   // 32768

static __device__ __forceinline__ v8f wmma_f32(v2f a, v2f b, v8f c) {
    // D = A(16x4 f32) * B(4x16 f32) + C(16x16 f32); wave32, EXEC must be all 1s.
    return __builtin_amdgcn_wmma_f32_16x16x4_f32(false, a, false, b, (short)0, c, false, false);
}

// ---------------- Kernel 1: h = relu(z @ W1 + b1), h is [64,256] ----------------
__global__ __launch_bounds__(256)
void k_gemm1_relu(const float* __restrict__ z, const float* __restrict__ W1,
                  const float* __restrict__ b1, float* __restrict__ h)
{
    int lane  = threadIdx.x & 31;
    int wid   = threadIdx.x >> 5;
    int tile  = blockIdx.x * 8 + wid;   // 64 tiles total (4 M x 16 N)
    int mtile = tile & 3;
    int ntile = tile >> 2;
    int row   = lane & 15;
    int half  = lane >> 4;

    v8f acc = {};
    const float* zrow = z + (mtile * 16 + row) * LATENT;
    for (int k0 = 0; k0 < LATENT; k0 += 4) {
        int kb = k0 + 2 * half;
        float2 t = *(const float2*)(zrow + kb);      // A: K = kb, kb+1
        v2f a; a.x = t.x; a.y = t.y;
        v2f b;
        b.x = W1[(kb + 0) * HID + ntile * 16 + row]; // B: lane = N, VGPR = K
        b.y = W1[(kb + 1) * HID + ntile * 16 + row];
        acc = wmma_f32(a, b, acc);
    }
    int n = ntile * 16 + row;
    float bias = b1[n];
    #pragma unroll
    for (int v = 0; v < 8; ++v) {
        int m = mtile * 16 + v + 8 * half;
        float val = acc[v] + bias;
        h[m * HID + n] = val > 0.f ? val : 0.f;
    }
}

// --------- Kernel 2: node_feats = h @ W2 + b2, [64,32768]; h staged in LDS ---------
__global__ __launch_bounds__(256)
void k_gemm2(const float* __restrict__ h, const float* __restrict__ W2,
             const float* __restrict__ b2, float* __restrict__ nf)
{
    __shared__ float hs[BATCH * HID];   // 64 KB of 320 KB WGP LDS
    {
        const float4* src = (const float4*)h;
        float4*       dst = (float4*)hs;
        for (int i = threadIdx.x; i < (BATCH * HID) / 4; i += 256) dst[i] = src[i];
    }
    __syncthreads();

    int lane  = threadIdx.x & 31;
    int wid   = threadIdx.x >> 5;
    int ntile = blockIdx.x * 8 + wid;   // 2048 N-tiles of width 16
    int col   = lane & 15;
    int half  = lane >> 4;
    int n     = ntile * 16 + col;

    v8f acc[4] = {};                    // four 16-row M tiles -> full M=64
    for (int k0 = 0; k0 < HID; k0 += 4) {
        int kb = k0 + 2 * half;
        v2f b;
        b.x = W2[(kb + 0) * NF_N + n];  // W2 panel read exactly once device-wide
        b.y = W2[(kb + 1) * NF_N + n];
        #pragma unroll
        for (int mt = 0; mt < 4; ++mt) {
            float2 t = *(const float2*)(hs + (mt * 16 + col) * HID + kb);
            v2f a; a.x = t.x; a.y = t.y;
            acc[mt] = wmma_f32(a, b, acc[mt]);
        }
    }
    float bias = b2[n];
    #pragma unroll
    for (int mt = 0; mt < 4; ++mt) {
        #pragma unroll
        for (int v = 0; v < 8; ++v) {
            int m = mt * 16 + v + 8 * half;
            nf[m * NF_N + n] = acc[mt][v] + bias;
        }
    }
}

// ---------------- Kernel 3: si/sj = node_feats . wi/wj, each [64,512] ----------------
__global__ __launch_bounds__(256)
void k_scores(const float* __restrict__ nf, const float* __restrict__ We,
              float* __restrict__ si, float* __restrict__ sj)
{
    int idx = blockIdx.x * 256 + threadIdx.x;   // 0 .. 64*512-1
    const float4* p  = (const float4*)(nf + (size_t)idx * NODE_FEAT);
    const float4* wi = (const float4*)We;
    const float4* wj = (const float4*)(We + NODE_FEAT);
    float ai = 0.f, aj = 0.f;
    #pragma unroll
    for (int q = 0; q < NODE_FEAT / 4; ++q) {
        float4 v = p[q];
        float4 a = wi[q];
        float4 b = wj[q];
        ai += v.x * a.x + v.y * a.y + v.z * a.z + v.w * a.w;
        aj += v.x * b.x + v.y * b.y + v.z * b.z + v.w * b.w;
    }
    si[idx] = ai;
    sj[idx] = aj;
}

// ---------------- Kernel 4: A_hat = sigmoid(si + sj + be), 67 MB stream ----------------
__global__ __launch_bounds__(256)
void k_edges(const float* __restrict__ si, const float* __restrict__ sj,
             const float* __restrict__ be, float* __restrict__ A)
{
    size_t e  = ((size_t)blockIdx.x * 256 + threadIdx.x) * 4;
    int b     = (int)(e >> 18);          // / (512*512)
    int rem   = (int)(e & 262143);
    int ni    = rem >> 9;
    int nj    = rem & 511;               // multiple of 4
    float s   = si[b * MAX_NODES + ni] + be[0];
    float4 sv = *(const float4*)(sj + b * MAX_NODES + nj);
    float4 o;
    o.x = 1.f / (1.f + __expf(-(s + sv.x)));
    o.y = 1.f / (1.f + __expf(-(s + sv.y)));
    o.z = 1.f / (1.f + __expf(-(s + sv.z)));
    o.w = 1.f / (1.f + __expf(-(s + sv.w)));
    *(float4*)(A + e) = o;
}

extern "C" void kernel_launch(void* const* d_in, const int* in_sizes, int n_in,
                              void* d_out, int out_size, void* d_ws, size_t ws_size,
                              hipStream_t stream) {
    const float* z  = (const float*)d_in[0];
    const float* W1 = (const float*)d_in[1];
    const float* b1 = (const float*)d_in[2];
    const float* W2 = (const float*)d_in[3];
    const float* b2 = (const float*)d_in[4];
    const float* We = (const float*)d_in[5];
    const float* be = (const float*)d_in[6];

    float* out = (float*)d_out;
    float* nf  = out;                              // node_feats [64,512,64]
    float* Ah  = out + (size_t)BATCH * NF_N;       // A_hat [64,512,512]

    float* h  = (float*)d_ws;                      // [64,256]
    float* si = h + BATCH * HID;                   // [64,512]
    float* sj = si + BATCH * MAX_NODES;            // [64,512]

    k_gemm1_relu<<<8, 256, 0, stream>>>(z, W1, b1, h);
    k_gemm2<<<256, 256, 0, stream>>>(h, W2, b2, nf);
    k_scores<<<(BATCH * MAX_NODES) / 256, 256, 0, stream>>>(nf, We, si, sj);
    k_edges<<<(int)(((size_t)BATCH * MAX_NODES * MAX_NODES) / (256 * 4)), 256, 0, stream>>>(si, sj, be, Ah);
}